// GNN_v1_37649683317088
// MI455X (gfx1250) — compile-verified
//
#include <hip/hip_runtime.h>
#include <hip/hip_bf16.h>

typedef float v2f __attribute__((ext_vector_type(2)));
typedef float v8f __attribute__((ext_vector_type(8)));

#define D 128
#define GNUM 64
#define HGDIM 257   // 2*128 + 1

// ---------------- elementwise init kernels ----------------
__global__ void k_fill(float* p, float v, int n) {
    int i = blockIdx.x * blockDim.x + threadIdx.x;
    if (i < n) p[i] = v;
}

// ---------------- degree accumulation ----------------
__global__ void k_deg_edges(const int* __restrict__ ei, const float* __restrict__ ew,
                            float* __restrict__ deg, int E) {
    int e = blockIdx.x * blockDim.x + threadIdx.x;
    if (e >= E) return;
    int dst = ei[E + e];
    atomicAdd(&deg[dst], ew[e]);
}

__global__ void k_dinv(const float* __restrict__ deg, float* __restrict__ dinv, int N) {
    int i = blockIdx.x * blockDim.x + threadIdx.x;
    if (i >= N) return;
    float d = deg[i];
    dinv[i] = (d > 0.0f) ? rsqrtf(d) : 0.0f;
}

// ---------------- per-edge normalization coefficient ----------------
__global__ void k_norm(const int* __restrict__ ei, const float* __restrict__ ew,
                       const float* __restrict__ dinv, float* __restrict__ nrm, int E) {
    int e = blockIdx.x * blockDim.x + threadIdx.x;
    if (e >= E) return;
    int s = ei[e];
    int t = ei[E + e];
    nrm[e] = dinv[s] * ew[e] * dinv[t];
}

// ---------------- fp32 WMMA GEMM: hW[N,128] = X[N,128] @ W[128,128] ----------------
// One block = 16 rows of X; 8 waves, wave w computes the 16x16 tile at cols [16w,16w+16).
// A tile (16x128) staged in LDS, shared by all 8 waves.
__global__ void __launch_bounds__(256)
k_gemm_xw(const float* __restrict__ X, const float* __restrict__ W,
          float* __restrict__ Y) {
    __shared__ float As[16 * D];   // 8 KB

    const int t    = threadIdx.x;        // 0..255
    const int lane = t & 31;
    const int wave = t >> 5;             // 0..7 -> col tile
    const int m0   = blockIdx.x * 16;
    const int n0   = wave * 16;
    const int row  = lane & 15;          // M (for A), N (for B/C)
    const int half = lane >> 4;          // 0/1

    // cooperative load of the 16x128 A tile (2048 floats, 8 per thread)
    {
        const float* src = X + (size_t)m0 * D;
        #pragma unroll
        for (int i = 0; i < 8; ++i)
            As[t * 8 + i] = src[t * 8 + i];
    }
    __syncthreads();

    v8f c = {};
    #pragma unroll 4
    for (int k0 = 0; k0 < D; k0 += 4) {
        const int kk = k0 + 2 * half;
        v2f a, b;
        // A 16x4 frag: VGPR j = A[M=row][K = 2*half + j]
        a.x = As[row * D + kk + 0];
        a.y = As[row * D + kk + 1];
        // B 4x16 frag: VGPR j = W[K = 2*half + j][N = n0 + row]
        b.x = W[(kk + 0) * D + n0 + row];
        b.y = W[(kk + 1) * D + n0 + row];
        c = __builtin_amdgcn_wmma_f32_16x16x4_f32(
                /*neg_a=*/false, a, /*neg_b=*/false, b,
                /*c_mod=*/(short)0, c, /*reuse_a=*/false, /*reuse_b=*/false);
    }

    // C/D: VGPR j holds M = j + 8*half, N = n0 + row
    #pragma unroll
    for (int j = 0; j < 8; ++j)
        Y[(size_t)(m0 + j + 8 * half) * D + n0 + row] = c[j];
}

// ---------------- edge scatter: agg[dst] += hW[src] * norm ----------------
// One wave (32 lanes) per edge, float4 per lane (32*4 = 128 features).
__global__ void k_scatter(const int* __restrict__ ei, const float* __restrict__ nrm,
                          const float* __restrict__ hW, float* __restrict__ agg, int E) {
    int idx = blockIdx.x * blockDim.x + threadIdx.x;
    int e = idx >> 5;                 // 32 lanes per edge
    int q = (idx & 31) * 4;           // feature base
    if (e >= E) return;
    int s = ei[e];                    // wave-uniform loads
    int t = ei[E + e];
    float w = nrm[e];
    const float4 hv = *(const float4*)(hW + (size_t)s * D + q);   // global_load_b128
    float* ap = agg + (size_t)t * D + q;
    atomicAdd(ap + 0, hv.x * w);
    atomicAdd(ap + 1, hv.y * w);
    atomicAdd(ap + 2, hv.z * w);
    atomicAdd(ap + 3, hv.w * w);
}

// ---------------- add self-loop term + bias, ReLU (in place on agg) ----------------
__global__ void k_finish(const float* __restrict__ hW, float* __restrict__ agg,
                         const float* __restrict__ dinv, const float* __restrict__ cb,
                         int N) {
    int idx = blockIdx.x * blockDim.x + threadIdx.x;
    int n = idx >> 7;
    int d = idx & (D - 1);
    if (n >= N) return;
    float di = dinv[n];
    float v = agg[(size_t)n * D + d] + hW[(size_t)n * D + d] * (di * di) + cb[d];
    agg[(size_t)n * D + d] = fmaxf(v, 0.0f);
}

// ---------------- segment max / sum / count pooling ----------------
__global__ void k_pool(const float* __restrict__ h, const int* __restrict__ batch,
                       float* __restrict__ gmax, float* __restrict__ gsum,
                       float* __restrict__ cnt, int N) {
    int idx = blockIdx.x * blockDim.x + threadIdx.x;
    int n = idx >> 7;
    int d = idx & (D - 1);
    if (n >= N) return;
    int g = batch[n];
    float v = h[(size_t)n * D + d];
    // v >= 0 (post-ReLU) and init is 0.0f, so uint-bit max == float max
    atomicMax((unsigned int*)&gmax[g * D + d], __float_as_uint(v));
    atomicAdd(&gsum[g * D + d], v);
    if (d == 0) atomicAdd(&cnt[g], 1.0f);
}

// ---------------- build hg = [max | mean | rho] ----------------
__global__ void k_build_hg(const float* __restrict__ gmax, const float* __restrict__ gsum,
                           const float* __restrict__ cnt, const float* __restrict__ rho,
                           float* __restrict__ hg) {
    int idx = blockIdx.x * blockDim.x + threadIdx.x;
    if (idx >= GNUM * HGDIM) return;
    int g = idx / HGDIM, c = idx % HGDIM;
    float v;
    if (c < D)            v = gmax[g * D + c];
    else if (c < 2 * D)   v = gsum[g * D + (c - D)] / fmaxf(cnt[g], 1.0f);
    else                  v = rho[g];
    hg[idx] = v;
}

// ---------------- tiny dense layers ----------------
__global__ void k_lin(const float* __restrict__ A, const float* __restrict__ W,
                      const float* __restrict__ b, float* __restrict__ Z,
                      int M, int K, int Nc, int do_relu) {
    int idx = blockIdx.x * blockDim.x + threadIdx.x;
    if (idx >= M * Nc) return;
    int m = idx / Nc, n = idx % Nc;
    float acc = b[n];
    for (int k = 0; k < K; ++k)
        acc = fmaf(A[m * K + k], W[k * Nc + n], acc);
    Z[idx] = do_relu ? fmaxf(acc, 0.0f) : acc;
}

extern "C" void kernel_launch(void* const* d_in, const int* in_sizes, int n_in,
                              void* d_out, int out_size, void* d_ws, size_t ws_size,
                              hipStream_t stream) {
    const float* x   = (const float*)d_in[0];
    const float* ew  = (const float*)d_in[1];
    const float* rho = (const float*)d_in[2];
    const float* cw  = (const float*)d_in[3];
    const float* cb  = (const float*)d_in[4];
    const float* w1  = (const float*)d_in[5];
    const float* b1  = (const float*)d_in[6];
    const float* w2  = (const float*)d_in[7];
    const float* b2  = (const float*)d_in[8];
    const float* w3  = (const float*)d_in[9];
    const float* b3  = (const float*)d_in[10];
    const int*   ei  = (const int*)d_in[11];
    const int*   bat = (const int*)d_in[12];

    const int N = in_sizes[0] / D;     // 40000
    const int E = in_sizes[1];         // 640000

    // workspace carve-up
    char* ws = (char*)d_ws;
    float* deg  = (float*)ws; ws += (size_t)N * sizeof(float);
    float* dinv = (float*)ws; ws += (size_t)N * sizeof(float);
    float* nrm  = (float*)ws; ws += (size_t)E * sizeof(float);
    float* hW   = (float*)ws; ws += (size_t)N * D * sizeof(float);
    float* agg  = (float*)ws; ws += (size_t)N * D * sizeof(float);
    float* gmax = (float*)ws; ws += (size_t)GNUM * D * sizeof(float);
    float* gsum = (float*)ws; ws += (size_t)GNUM * D * sizeof(float);
    float* cnt  = (float*)ws; ws += (size_t)GNUM * sizeof(float);
    float* hg   = (float*)ws; ws += (size_t)GNUM * HGDIM * sizeof(float);
    float* z1   = (float*)ws; ws += (size_t)GNUM * D * sizeof(float);
    float* z2   = (float*)ws; ws += (size_t)GNUM * D * sizeof(float);

    const int B = 256;
    const int ND = N * D;                                  // 5,120,000
    const int poolN = GNUM * D * 2 + GNUM;                 // gmax|gsum|cnt contiguous
    const int scatterThreads = E * 32;                     // 20,480,000 (fits int)

    // init (re-done every call for graph-replay determinism)
    k_fill<<<(N + B - 1) / B, B, 0, stream>>>(deg, 1.0f, N);   // self-loop weight
    k_fill<<<(ND + B - 1) / B, B, 0, stream>>>(agg, 0.0f, ND);
    k_fill<<<(poolN + B - 1) / B, B, 0, stream>>>(gmax, 0.0f, poolN);

    // degree + norm
    k_deg_edges<<<(E + B - 1) / B, B, 0, stream>>>(ei, ew, deg, E);
    k_dinv<<<(N + B - 1) / B, B, 0, stream>>>(deg, dinv, N);
    k_norm<<<(E + B - 1) / B, B, 0, stream>>>(ei, ew, dinv, nrm, E);

    // hW = x @ conv_w  (fp32 WMMA)
    k_gemm_xw<<<N / 16, B, 0, stream>>>(x, cw, hW);

    // message passing (one wave per edge, float4 lanes)
    k_scatter<<<(scatterThreads + B - 1) / B, B, 0, stream>>>(ei, nrm, hW, agg, E);
    k_finish<<<(ND + B - 1) / B, B, 0, stream>>>(hW, agg, dinv, cb, N);

    // pooling
    k_pool<<<(ND + B - 1) / B, B, 0, stream>>>(agg, bat, gmax, gsum, cnt, N);
    k_build_hg<<<(GNUM * HGDIM + B - 1) / B, B, 0, stream>>>(gmax, gsum, cnt, rho, hg);

    // MLP head
    k_lin<<<(GNUM * D + B - 1) / B, B, 0, stream>>>(hg, w1, b1, z1, GNUM, HGDIM, D, 1);
    k_lin<<<(GNUM * D + B - 1) / B, B, 0, stream>>>(z1, w2, b2, z2, GNUM, D, D, 1);
    k_lin<<<(GNUM * 36 + B - 1) / B, B, 0, stream>>>(z2, w3, b3, (float*)d_out, GNUM, D, 36, 0);
}